// QuadraticSplineRenderer_50886772523415
// MI455X (gfx1250) — compile-verified
//
#include <hip/hip_runtime.h>
#include <math.h>

// ---------------------------------------------------------------------------
// QuadraticSplineRenderer for MI455X (gfx1250, wave32)
//   - per-64x64-tile curve culling (matches reference _bbox_mask semantics)
//   - per-curve invariants precomputed once per block into LDS
//   - curve data staged via CDNA5 async global->LDS DMA (ASYNCcnt path)
// ---------------------------------------------------------------------------

#define IMG       512
#define TILE      64
#define NSPL      128
#define SLICES    4          // blocks per tile (each does 16 rows x 64 cols)
#define THREADS   256        // 8 wave32 waves
#define PSTRIDE   12         // floats per precomputed curve record

#if defined(__gfx1250__) && __has_builtin(__builtin_amdgcn_global_load_async_to_lds_b32)
#define ASYNC_LDS 1
#else
#define ASYNC_LDS 0
#endif

#if ASYNC_LDS
typedef __attribute__((address_space(1))) int* gptr_i32;
typedef __attribute__((address_space(3))) int* lptr_i32;

__device__ __forceinline__ void async_cp_b32(const float* g, float* l) {
  __builtin_amdgcn_global_load_async_to_lds_b32(
      (gptr_i32)const_cast<float*>(g),
      (lptr_i32)l,
      /*offset=*/0, /*cpol=*/0);
}
__device__ __forceinline__ void wait_async() {
#if __has_builtin(__builtin_amdgcn_s_wait_asynccnt)
  __builtin_amdgcn_s_wait_asynccnt(0);
#else
  asm volatile("s_wait_asynccnt 0" ::: "memory");
#endif
}
#endif

__global__ __launch_bounds__(THREADS)
void qspline_render_kernel(const float* __restrict__ ga,   // (128,2) y,x
                           const float* __restrict__ gb,
                           const float* __restrict__ gc,
                           const float* __restrict__ glw,  // (128)
                           const float* __restrict__ glc,  // (128)
                           float* __restrict__ out)        // (512,512)
{
  __shared__ float s_raw[1024];            // a(256) b(256) c(256) lw(128) lc(128)
  __shared__ float s_p[NSPL * PSTRIDE];    // precomputed per-curve params
  __shared__ int   s_idx[NSPL];            // flags, then compacted hit list
  __shared__ int   s_cnt;

  const int tid   = threadIdx.x;
  const int blk   = blockIdx.x;
  const int tile  = blk >> 2;              // 0..63
  const int slice = blk & 3;               // 0..3  (16-row slice of the tile)
  const int tiy   = tile >> 3;             // tile row 0..7
  const int tix   = tile & 7;              // tile col 0..7

  // ---- stage curve data into LDS (async DMA on gfx1250) -------------------
#if ASYNC_LDS
  async_cp_b32(ga + tid, s_raw + tid);
  async_cp_b32(gb + tid, s_raw + 256 + tid);
  async_cp_b32(gc + tid, s_raw + 512 + tid);
  if (tid < NSPL) {
    async_cp_b32(glw + tid, s_raw + 768 + tid);
    async_cp_b32(glc + tid, s_raw + 896 + tid);
  }
  wait_async();
#else
  s_raw[tid]       = ga[tid];
  s_raw[256 + tid] = gb[tid];
  s_raw[512 + tid] = gc[tid];
  if (tid < NSPL) {
    s_raw[768 + tid] = glw[tid];
    s_raw[896 + tid] = glc[tid];
  }
#endif
  __syncthreads();

  // ---- phase 1: per-curve tile mask + invariant precompute ----------------
  if (tid < NSPL) {
    const float ay = s_raw[2 * tid],       ax = s_raw[2 * tid + 1];
    const float by = s_raw[256 + 2 * tid], bx = s_raw[256 + 2 * tid + 1];
    const float cy = s_raw[512 + 2 * tid], cx = s_raw[512 + 2 * tid + 1];
    const float lw = s_raw[768 + tid];
    const float lc = s_raw[896 + tid];

    // _bbox_mask (uses raw b, not b+1e-5)
    float miy = fminf(ay, cy), may = fmaxf(ay, cy);
    float mix = fminf(ax, cx), maxx = fmaxf(ax, cx);
    const bool cond = (by < miy) || (by > may) || (bx < mix) || (bx > maxx);
    const float dny = ay - 2.0f * by + cy;
    const float dnx = ax - 2.0f * bx + cx;
    const bool dok = (fabsf(dny) > 1e-9f) && (fabsf(dnx) > 1e-9f);
    float ty_ = (ay - by) / (fabsf(dny) > 1e-9f ? dny : 1e-9f);
    float tx_ = (ax - bx) / (fabsf(dnx) > 1e-9f ? dnx : 1e-9f);
    ty_ = fminf(fmaxf(ty_, 0.0f), 1.0f);
    tx_ = fminf(fmaxf(tx_, 0.0f), 1.0f);
    const float sy = 1.0f - ty_, sx = 1.0f - tx_;
    const float qy = sy * sy * ay + 2.0f * sy * ty_ * by + ty_ * ty_ * cy;
    const float qx = sx * sx * ax + 2.0f * sx * tx_ * bx + tx_ * tx_ * cx;
    if (cond && dok) {
      miy = fminf(miy, qy); may = fmaxf(may, qy);
      mix = fminf(mix, qx); maxx = fmaxf(maxx, qx);
    }
    const float mg   = floorf(3.0f * lw);                 // MARGIN == 0
    const float miyp = fminf(fmaxf(floorf(miy * 512.0f) - mg, 0.0f), 512.0f);
    const float mayp = fminf(fmaxf(ceilf (may * 512.0f) + mg, 0.0f), 512.0f);
    const float mixp = fminf(fmaxf(floorf(mix * 512.0f) - mg, 0.0f), 512.0f);
    const float maxp = fminf(fmaxf(ceilf (maxx * 512.0f) + mg, 0.0f), 512.0f);
    const float t0y = (float)(tiy * TILE), t0x = (float)(tix * TILE);
    const bool hit = (mayp > t0y) && (miyp < t0y + 64.0f) &&
                     (maxp > t0x) && (mixp < t0x + 64.0f);
    s_idx[tid] = hit ? 1 : 0;

    // forward-pass invariants (use bb = b + 1e-5)
    const float bby = by + 1e-5f, bbx = bx + 1e-5f;
    const float Ay = bby - ay, Ax = bbx - ax;
    const float By = cy - bby - Ay, Bx = cx - bbx - Ax;      // Bq
    const float BdB  = By * By + Bx * Bx;
    const float invB = 1.0f / BdB;                           // note: /-BdB folded into signs
    float* P = &s_p[tid * PSTRIDE];
    P[0] = ay;  P[1] = ax;  P[2] = Ay;  P[3] = Ax;
    P[4] = By;  P[5] = Bx;  P[6] = invB;
    P[7] = 3.0f * (Ay * By + Ax * Bx) * invB;                // ca
    P[8] = 2.0f * (Ay * Ay + Ax * Ax);                       // 2*A.A
    P[9] = 6.0f / lw;                                        // sigmoid slope
    P[10] = lc; P[11] = 0.0f;
  }
  __syncthreads();

  if (tid == 0) {                    // deterministic serial compaction
    int c = 0;
    for (int i = 0; i < NSPL; ++i)
      if (s_idx[i]) s_idx[c++] = i;  // c <= i, safe in place
    s_cnt = c;
  }
  __syncthreads();

  // ---- phase 2: 4 pixels per thread, curves outer for param reuse ---------
  const int   row0 = tiy * TILE + slice * 16 + (tid >> 6);   // +4*it below
  const int   col  = tix * TILE + (tid & 63);
  const float stepv = 1.0f / 511.0f;                          // linspace(0,1,512)
  const float pxf   = (float)col * stepv;
  const int   cnt   = s_cnt;

  float acc[4] = {0.0f, 0.0f, 0.0f, 0.0f};

  for (int j = 0; j < cnt; ++j) {
    const float* P = &s_p[s_idx[j] * PSTRIDE];               // LDS broadcast
    const float ay = P[0], ax = P[1], Ay = P[2], Ax = P[3];
    const float By = P[4], Bx = P[5], invB = P[6], ca = P[7];
    const float twoAA = P[8], kk = P[9], lcv = P[10];

    const float Cx  = pxf - ax;
    const float cBx = Cx * Bx;
    const float cAx = Cx * Ax;
    const float ca3 = ca * (1.0f / 3.0f);

#pragma unroll
    for (int it = 0; it < 4; ++it) {
      const float pyf = (float)(row0 + 4 * it) * stepv;
      const float Cy  = pyf - ay;
      const float cb  = (twoAA - (Cy * By + cBx)) * invB;    // == (C.B - 2A.A)/-BdB
      const float cc  = -(Cy * Ay + cAx) * invB;             // == (C.A)/-BdB

      // ---- closed-form cubic (matches _solve_cubic) ----
      const float pp  = cb - ca * ca3;
      const float p3  = pp * pp * pp;
      const float qq  = ca * (2.0f * ca * ca - 9.0f * cb) * (1.0f / 27.0f) + cc;
      const float dd  = qq * qq + 4.0f * p3 * (1.0f / 27.0f);
      const float sd  = sqrtf(fmaxf(dd, 0.0f));
      const float r0  = cbrtf((sd - qq) * 0.5f) + cbrtf((-sd - qq) * 0.5f) - ca3;
      const float p3s = (fabsf(p3) > 1e-9f) ? p3 : 1e-9f;
      float arg = -sqrtf(fmaxf(-27.0f / p3s, 0.0f)) * qq * 0.5f;
      arg = fminf(fmaxf(arg, -1.0f), 1.0f);
      const float v  = acosf(arg) * (1.0f / 3.0f);
      const float m  = __cosf(v);
      const float n  = __sinf(v) * 1.7320508075688772f;
      const float sp = sqrtf(fmaxf(-pp * (1.0f / 3.0f), 0.0f));
      const float r1 = (m + m) * sp - ca3;
      const float r2 = (-n - m) * sp - ca3;

      float t0 = (dd > 0.0f) ? r0 : r1;
      float t1 = (dd > 0.0f) ? r0 : r2;
      t0 = fminf(fmaxf(t0, 0.0f), 1.0f);
      t1 = fminf(fmaxf(t1, 0.0f), 1.0f);

      const float d1y = (2.0f * Ay + By * t0) * t0 - Cy;
      const float d1x = (2.0f * Ax + Bx * t0) * t0 - Cx;
      const float d2y = (2.0f * Ay + By * t1) * t1 - Cy;
      const float d2x = (2.0f * Ax + Bx * t1) * t1 - Cx;
      const float dsq = fminf(d1y * d1y + d1x * d1x, d2y * d2y + d2x * d2x);
      const float dist = sqrtf(dsq);

      // sigmoid((lw-dist)/lw*6)*lc == lc / (1 + exp(kk*dist - 6))
      acc[it] += lcv / (1.0f + __expf(kk * dist - 6.0f));
    }
  }

#pragma unroll
  for (int it = 0; it < 4; ++it)
    out[(row0 + 4 * it) * IMG + col] = 1.0f - acc[it];
}

extern "C" void kernel_launch(void* const* d_in, const int* in_sizes, int n_in,
                              void* d_out, int out_size, void* d_ws, size_t ws_size,
                              hipStream_t stream) {
  (void)in_sizes; (void)n_in; (void)out_size; (void)d_ws; (void)ws_size;
  const float* a  = (const float*)d_in[0];
  const float* b  = (const float*)d_in[1];
  const float* c  = (const float*)d_in[2];
  const float* lw = (const float*)d_in[3];
  const float* lc = (const float*)d_in[4];
  float* out = (float*)d_out;

  const int nblocks = (IMG / TILE) * (IMG / TILE) * SLICES;   // 64 tiles * 4 = 256
  qspline_render_kernel<<<nblocks, THREADS, 0, stream>>>(a, b, c, lw, lc, out);
}